// DispaxD3_78735340470316
// MI455X (gfx1250) — compile-verified
//
#include <hip/hip_runtime.h>
#include <math.h>

typedef __attribute__((ext_vector_type(2))) float v2f;
typedef __attribute__((ext_vector_type(8))) float v8f;

#define BOHR_F 0.5291772105638411f
#define HA_F   27.211386024367243f
#define S6_F   1.0f
#define S8_F   0.7875f
#define A1_F   0.4289f
#define A2_F   4.4407f
#define KCN_F  16.0f
#define WF_F   4.0f
#define NREF   5
#define WPAD   8     // padded weights row stride (floats)
#define CPAD   32    // padded ref_c6 row stride (floats) -> 128B aligned rows
#define NBINS  64
#define EPS_F  1.1920929e-07f

__device__ __forceinline__ float smooth_cutoff(float dr, float r_on, float r_cut) {
    float r_c = r_cut * r_cut;
    float r_o = r_on * r_on;
    float r   = dr * dr;
    float den = r_c - r_o;
    float t   = r_c - r;
    float inner = (dr < r_cut) ? (t * t * (r_c + 2.0f * r - 3.0f * r_o) / (den * den * den))
                               : 0.0f;
    return (dr < r_on) ? 1.0f : inner;
}

// Init: zero cn[] and the 64 energy bins
__global__ void k_init(float* __restrict__ cn, int n, float* __restrict__ bins) {
    int a = blockIdx.x * blockDim.x + threadIdx.x;
    if (a < n) cn[a] = 0.0f;
    if (a < NBINS) bins[a] = 0.0f;
}

// Repack ref_c6 [NELEM,NELEM,5,5] (100B rows) -> c6pad (128B rows, b128-loadable)
__global__ void k_pack_c6(const float* __restrict__ ref_c6, float* __restrict__ c6pad, int npair) {
    int t = blockIdx.x * blockDim.x + threadIdx.x;
    int pair = t / (NREF * NREF);
    int r    = t % (NREF * NREF);
    if (pair < npair) c6pad[(size_t)pair * CPAD + r] = ref_c6[(size_t)pair * (NREF * NREF) + r];
}

// Pass 1: per-edge coordination-number contribution, scatter-add into cn[i]
__global__ void k_edge_cn(const float* __restrict__ drv, const int* __restrict__ idx,
                          const int* __restrict__ numbers, const float* __restrict__ rcov,
                          float* __restrict__ cn, int E) {
    int e = blockIdx.x * blockDim.x + threadIdx.x;
    if (e >= E) return;
    float x = drv[3 * e + 0];
    float y = drv[3 * e + 1];
    float z = drv[3 * e + 2];
    float dr = sqrtf(x * x + y * y + z * z) * (1.0f / BOHR_F);
    int i = idx[e];
    int j = idx[E + e];
    int Zi = numbers[i];
    int Zj = numbers[j];
    float rc = rcov[Zi] + rcov[Zj];
    float count = 1.0f / (1.0f + expf(-KCN_F * (rc / dr - 1.0f)));
    float masked = (dr > 0.0f) ? smooth_cutoff(dr, 20.0f, 25.0f) * count : 0.0f;
    atomicAdd(&cn[i], masked);
}

// Pass 2: per-atom normalized Gaussian weights -> padded rows of WPAD floats
__global__ void k_atom_w(const float* __restrict__ cn, const int* __restrict__ numbers,
                         const float* __restrict__ ref_cn, float* __restrict__ wpad, int n) {
    int a = blockIdx.x * blockDim.x + threadIdx.x;
    if (a >= n) return;
    float c = cn[a];
    int Z = numbers[a];
    float w[NREF];
    float s = 0.0f;
#pragma unroll
    for (int r = 0; r < NREF; ++r) {
        float rcn = ref_cn[Z * NREF + r];
        float d = rcn - c;
        float wv = (rcn >= 0.0f) ? expf(-WF_F * d * d) : 0.0f;
        w[r] = wv;
        s += wv;
    }
    float inv = 1.0f / (s + EPS_F);
#pragma unroll
    for (int r = 0; r < NREF; ++r) wpad[(size_t)a * WPAD + r] = w[r] * inv;
}

// Pass 3: per-edge BJ-damped dispersion energy; block sum -> atomicAdd into 64 bins
#define EPT 8
__global__ __launch_bounds__(256)
void k_edge_e(const float* __restrict__ drv, const int* __restrict__ idx,
              const int* __restrict__ numbers, const float* __restrict__ r4r2,
              const float* __restrict__ wpad, const float* __restrict__ c6pad,
              float* __restrict__ bins, int E, int nelem) {
    __shared__ float sred[256];
    float acc = 0.0f;
    int base = blockIdx.x * blockDim.x * EPT;
#pragma unroll
    for (int t = 0; t < EPT; ++t) {
        int e = base + t * blockDim.x + threadIdx.x;
        if (e < E) {
            int i = idx[e];
            int j = idx[E + e];
            int Zi = numbers[i];
            int Zj = numbers[j];
            const float* c6p = c6pad + (size_t)(Zj * nelem + Zi) * CPAD;
            __builtin_prefetch(c6p, 0, 1);  // global_prefetch_b8

            float x = drv[3 * e + 0];
            float y = drv[3 * e + 1];
            float z = drv[3 * e + 2];
            float dr = sqrtf(x * x + y * y + z * z) * (1.0f / BOHR_F);

            float qq = 3.0f * r4r2[Zi] * r4r2[Zj];

            // weights: one b128 + one b32 gather per atom (rows are 32B-aligned)
            const float4* wip = (const float4*)(wpad + (size_t)i * WPAD);
            const float4* wjp = (const float4*)(wpad + (size_t)j * WPAD);
            float4 wi0 = wip[0];
            float4 wj0 = wjp[0];
            float wi[NREF] = {wi0.x, wi0.y, wi0.z, wi0.w, wpad[(size_t)i * WPAD + 4]};
            float wj[NREF] = {wj0.x, wj0.y, wj0.z, wj0.w, wpad[(size_t)j * WPAD + 4]};

            // ref_c6 tile: six b128 + one b32 (rows are 128B-aligned)
            const float4* p4 = (const float4*)c6p;
            float rc6[NREF * NREF];
#pragma unroll
            for (int q = 0; q < 6; ++q) {
                float4 v = p4[q];
                rc6[4 * q + 0] = v.x;
                rc6[4 * q + 1] = v.y;
                rc6[4 * q + 2] = v.z;
                rc6[4 * q + 3] = v.w;
            }
            rc6[24] = c6p[24];

            float c6 = 0.0f;
#pragma unroll
            for (int k = 0; k < NREF; ++k) {
#pragma unroll
                for (int l = 0; l < NREF; ++l) {
                    c6 += wj[k] * wi[l] * rc6[k * NREF + l];
                }
            }
            float rr  = A1_F * sqrtf(qq) + A2_F;
            float dr2 = dr * dr;
            float dr6 = dr2 * dr2 * dr2;
            float dr8 = dr6 * dr2;
            float rr2 = rr * rr;
            float rr6 = rr2 * rr2 * rr2;
            float rr8 = rr6 * rr2;
            float damped = -c6 * (S6_F / (dr6 + rr6) + S8_F * qq / (dr8 + rr8));
            acc += smooth_cutoff(dr, 55.0f, 60.0f) * damped * 0.5f;
        }
    }
    sred[threadIdx.x] = acc;
    __syncthreads();
    for (int off = 128; off > 0; off >>= 1) {
        if (threadIdx.x < off) sred[threadIdx.x] += sred[threadIdx.x + off];
        __syncthreads();
    }
    if (threadIdx.x == 0) atomicAdd(&bins[blockIdx.x & (NBINS - 1)], sred[0]);
}

// Pass 4: single-wave 64->1 reduction via V_WMMA_F32_16X16X4_F32.
// The 64 bins form a 16x4 f32 A-matrix (2 VGPRs/lane); B = ones(4x16);
// D[m][n] = rowsum(A,m). Lane L<16 holds D[0..7][L] in its 8 D-VGPRs,
// lane L>=16 holds D[8..15][L-16]; local sum + shfl_xor(16) = grand total.
__global__ void k_final(const float* __restrict__ bins, float* __restrict__ out) {
    int lane = threadIdx.x;  // 32 threads, one wave, EXEC all-1s
    float a0 = bins[lane];
    float a1 = bins[lane + 32];
    v2f A;
    A[0] = a0;
    A[1] = a1;
    v2f B;
    B[0] = 1.0f;
    B[1] = 1.0f;
    v8f C = {};
    v8f D = __builtin_amdgcn_wmma_f32_16x16x4_f32(
        /*neg_a=*/false, A, /*neg_b=*/false, B,
        /*c_mod=*/(short)0, C, /*reuse_a=*/false, /*reuse_b=*/false);
    float s = D[0] + D[1] + D[2] + D[3] + D[4] + D[5] + D[6] + D[7];
    s += __shfl_xor(s, 16, 32);
    if (lane == 0) out[0] = s * HA_F;
}

extern "C" void kernel_launch(void* const* d_in, const int* in_sizes, int n_in,
                              void* d_out, int out_size, void* d_ws, size_t ws_size,
                              hipStream_t stream) {
    const float* drv     = (const float*)d_in[0];  // [E,3]
    const float* ref_cn  = (const float*)d_in[1];  // [95,5]
    const float* ref_c6  = (const float*)d_in[2];  // [95,95,5,5]
    const float* r4r2    = (const float*)d_in[3];  // [95]
    const float* rcov    = (const float*)d_in[4];  // [95]
    const int*   numbers = (const int*)d_in[5];    // [N]
    const int*   idx     = (const int*)d_in[6];    // [2,E]

    const int E     = in_sizes[0] / 3;
    const int N     = in_sizes[5];
    const int NELEM = in_sizes[3];
    const int NPAIR = NELEM * NELEM;

    // ws layout (all 16B/128B aligned given N % 4 == 0):
    //   cn      : N floats
    //   wpad    : N * WPAD floats
    //   bins    : NBINS floats
    //   c6pad   : NPAIR * CPAD floats
    float* ws      = (float*)d_ws;
    float* cn      = ws;
    float* wpad    = ws + N;
    float* bins    = ws + N + (size_t)N * WPAD;
    float* c6pad   = bins + NBINS;

    float* out = (float*)d_out;

    const int TB = 256;
    k_init<<<(N + TB - 1) / TB, TB, 0, stream>>>(cn, N, bins);
    {
        int tot = NPAIR * NREF * NREF;
        k_pack_c6<<<(tot + TB - 1) / TB, TB, 0, stream>>>(ref_c6, c6pad, NPAIR);
    }
    k_edge_cn<<<(E + TB - 1) / TB, TB, 0, stream>>>(drv, idx, numbers, rcov, cn, E);
    k_atom_w<<<(N + TB - 1) / TB, TB, 0, stream>>>(cn, numbers, ref_cn, wpad, N);
    const int tile = TB * EPT;
    const int nb = (E + tile - 1) / tile;
    k_edge_e<<<nb, TB, 0, stream>>>(drv, idx, numbers, r4r2, wpad, c6pad, bins, E, NELEM);
    k_final<<<1, 32, 0, stream>>>(bins, out);
}